// PatchSampleF_35115652612732
// MI455X (gfx1250) — compile-verified
//
#include <hip/hip_runtime.h>
#include <math.h>

typedef __attribute__((ext_vector_type(2))) float v2f;
typedef __attribute__((ext_vector_type(8))) float v8f;

// Fused patch-gather + 2-layer MLP (hidden=out=256) + row L2-norm.
// One workgroup (8 waves, 256 threads) computes a 16-row tile of the
// [B*256, 256] output. All matmuls use V_WMMA_F32_16X16X4_F32.
//
//  rows   : g = b*256 + p  (p = patch id index), 16 rows per block, same b.
//  GEMM1  : xs[16,C] @ w1[C,256]  (+b1, relu)  -> h[16,256] (LDS)
//  GEMM2  : h[16,256] @ w2[256,256] (+b2)      -> y[16,256] (LDS)
//  norm   : out = y / (||y||_2 + 1e-7)
//
// WMMA f32 16x16x4 fragment layouts (ISA 7.12.2):
//  A (16x4): lane = (k>=2)*16 + m ; vgpr j holds K = 2*(lane>>4)+j
//  B (4x16): lane = (k>=2)*16 + n ; vgpr j holds K = 2*(lane>>4)+j
//  C/D     : vgpr r = row (halfsel*8 + r), lane&15 = col
#define LSTR 257  // LDS row stride (floats): 256 + 1 pad -> conflict-free

template <int C>
__global__ __launch_bounds__(256) void patch_mlp_kernel(
    const float* __restrict__ feat,   // [B, C, H, W]
    const int*   __restrict__ pid,    // [256]
    const float* __restrict__ w1,     // [C, 256]
    const float* __restrict__ b1,     // [256]
    const float* __restrict__ w2,     // [256, 256]
    const float* __restrict__ b2,     // [256]
    float*       __restrict__ out,    // [B*256, 256]
    int HW)                           // H*W
{
    __shared__ float bufA[16 * LSTR];   // xs tile, later reused for y
    __shared__ float bufB[16 * LSTR];   // h tile
    __shared__ float ssq[16 * 16];      // per-row partial sums of squares
    __shared__ float invn[16];          // per-row 1/(norm+eps)
    __shared__ int   pidrow[16];        // gathered patch ids for this tile

    const int tid     = threadIdx.x;
    const int lane    = tid & 31;
    const int wid     = tid >> 5;       // wave id 0..7
    const int rowbase = blockIdx.x * 16;      // global row of tile row 0
    const int batch   = rowbase >> 8;         // 16 | 256 -> uniform per block

    const float* fbase = feat + (size_t)batch * C * HW;

    if (tid < 16) pidrow[tid] = pid[(rowbase + tid) & 255];
    __syncthreads();

    // ---- gather xs tile: bufA[m][c] = feat[batch, c, pid] ----------------
    for (int idx = tid; idx < 16 * C; idx += 256) {
        int m = idx / C;
        int c = idx & (C - 1);
        bufA[m * LSTR + c] = fbase[(size_t)c * HW + pidrow[m]];
    }
    __syncthreads();

    const int halfsel = lane >> 4;      // 0: K even pair low, 1: high
    const int mrow    = lane & 15;      // A-fragment row / B,C,D column
    const int col0    = wid * 32 + mrow;        // first owned output column
    const int col1    = col0 + 16;              // second owned output column

    // ---- GEMM1: xs @ w1, bias + relu ------------------------------------
    v8f acc0 = {};
    v8f acc1 = {};
#pragma unroll 4
    for (int k0 = 0; k0 < C; k0 += 4) {
        int ka = k0 + 2 * halfsel;
        v2f a;
        a.x = bufA[mrow * LSTR + ka];
        a.y = bufA[mrow * LSTR + ka + 1];
        v2f bb0, bb1;
        bb0.x = w1[(size_t)ka * 256 + col0];
        bb0.y = w1[(size_t)(ka + 1) * 256 + col0];
        bb1.x = w1[(size_t)ka * 256 + col1];
        bb1.y = w1[(size_t)(ka + 1) * 256 + col1];
        acc0 = __builtin_amdgcn_wmma_f32_16x16x4_f32(
            false, a, false, bb0, (short)0, acc0, false, false);
        acc1 = __builtin_amdgcn_wmma_f32_16x16x4_f32(
            false, a, false, bb1, (short)0, acc1, false, false);
    }
    {
        float bias0 = b1[col0];
        float bias1 = b1[col1];
#pragma unroll
        for (int r = 0; r < 8; ++r) {
            int m = halfsel * 8 + r;
            bufB[m * LSTR + col0] = fmaxf(acc0[r] + bias0, 0.0f);
            bufB[m * LSTR + col1] = fmaxf(acc1[r] + bias1, 0.0f);
        }
    }
    __syncthreads();

    // ---- GEMM2: h @ w2, bias; y -> bufA ---------------------------------
    acc0 = (v8f){};
    acc1 = (v8f){};
#pragma unroll 4
    for (int k0 = 0; k0 < 256; k0 += 4) {
        int ka = k0 + 2 * halfsel;
        v2f a;
        a.x = bufB[mrow * LSTR + ka];
        a.y = bufB[mrow * LSTR + ka + 1];
        v2f bb0, bb1;
        bb0.x = w2[(size_t)ka * 256 + col0];
        bb0.y = w2[(size_t)(ka + 1) * 256 + col0];
        bb1.x = w2[(size_t)ka * 256 + col1];
        bb1.y = w2[(size_t)(ka + 1) * 256 + col1];
        acc0 = __builtin_amdgcn_wmma_f32_16x16x4_f32(
            false, a, false, bb0, (short)0, acc0, false, false);
        acc1 = __builtin_amdgcn_wmma_f32_16x16x4_f32(
            false, a, false, bb1, (short)0, acc1, false, false);
    }
    {
        float bias0 = b2[col0];
        float bias1 = b2[col1];
#pragma unroll
        for (int r = 0; r < 8; ++r) {
            int m = halfsel * 8 + r;
            bufA[m * LSTR + col0] = acc0[r] + bias0;
            bufA[m * LSTR + col1] = acc1[r] + bias1;
        }
    }
    __syncthreads();

    // ---- row-wise L2 norm -----------------------------------------------
    {
        int rrow = tid >> 4;            // 0..15
        int rj   = tid & 15;            // 0..15
        float s = 0.0f;
#pragma unroll
        for (int t = 0; t < 16; ++t) {
            float v = bufA[rrow * LSTR + rj + 16 * t];
            s += v * v;
        }
        ssq[rrow * 16 + rj] = s;
    }
    __syncthreads();
    if (tid < 16) {
        float s = 0.0f;
#pragma unroll
        for (int j = 0; j < 16; ++j) s += ssq[tid * 16 + j];
        invn[tid] = 1.0f / (sqrtf(s) + 1e-7f);
    }
    __syncthreads();

    for (int idx = tid; idx < 16 * 256; idx += 256) {
        int m = idx >> 8;
        int c = idx & 255;
        out[(size_t)(rowbase + m) * 256 + c] = bufA[m * LSTR + c] * invn[m];
    }
}

extern "C" void kernel_launch(void* const* d_in, const int* in_sizes, int n_in,
                              void* d_out, int out_size, void* d_ws, size_t ws_size,
                              hipStream_t stream) {
    (void)in_sizes; (void)n_in; (void)d_ws; (void)ws_size; (void)out_size;

    const float* feat0 = (const float*)d_in[0];   // [8,128,128,128]
    const float* feat1 = (const float*)d_in[1];   // [8,256,64,64]
    const float* feat2 = (const float*)d_in[2];   // [8,256,64,64]
    const int*   pid0  = (const int*)d_in[3];
    const int*   pid1  = (const int*)d_in[4];
    const int*   pid2  = (const int*)d_in[5];
    const float* w1_0 = (const float*)d_in[6];
    const float* b1_0 = (const float*)d_in[7];
    const float* w2_0 = (const float*)d_in[8];
    const float* b2_0 = (const float*)d_in[9];
    const float* w1_1 = (const float*)d_in[10];
    const float* b1_1 = (const float*)d_in[11];
    const float* w2_1 = (const float*)d_in[12];
    const float* b2_1 = (const float*)d_in[13];
    const float* w1_2 = (const float*)d_in[14];
    const float* b1_2 = (const float*)d_in[15];
    const float* w2_2 = (const float*)d_in[16];
    const float* b2_2 = (const float*)d_in[17];

    float* out = (float*)d_out;
    const int ROWS = 8 * 256;                 // 2048 rows per pipeline
    const int GRID = ROWS / 16;               // 128 workgroups

    patch_mlp_kernel<128><<<GRID, 256, 0, stream>>>(
        feat0, pid0, w1_0, b1_0, w2_0, b2_0, out, 128 * 128);
    patch_mlp_kernel<256><<<GRID, 256, 0, stream>>>(
        feat1, pid1, w1_1, b1_1, w2_1, b2_1, out + (size_t)ROWS * 256, 64 * 64);
    patch_mlp_kernel<256><<<GRID, 256, 0, stream>>>(
        feat2, pid2, w1_2, b1_2, w2_2, b2_2, out + (size_t)2 * ROWS * 256, 64 * 64);
}